// GNNModel_45131516346758
// MI455X (gfx1250) — compile-verified
//
#include <hip/hip_runtime.h>

typedef __attribute__((ext_vector_type(2))) float v2f;
typedef __attribute__((ext_vector_type(8))) float v8f;

#if defined(__has_builtin)
#if __has_builtin(__builtin_amdgcn_global_load_async_to_lds_b32)
#define HAVE_ASYNC_LDS 1
#endif
#endif
#ifndef HAVE_ASYNC_LDS
#define HAVE_ASYNC_LDS 0
#endif

typedef __attribute__((address_space(1))) int as1_int;   // global
typedef __attribute__((address_space(3))) int as3_int;   // LDS

// ---------------------------------------------------------------- degree prep
__global__ void k_init_deg(float* __restrict__ deg, int n) {
    int i = blockIdx.x * blockDim.x + threadIdx.x;
    if (i < n) deg[i] = 1.0f;   // self-loop weight
}

__global__ void k_accum_deg(float* __restrict__ deg,
                            const int* __restrict__ dst,
                            const float* __restrict__ w, int e) {
    int i = blockIdx.x * blockDim.x + threadIdx.x;
    if (i < e) atomicAdd(&deg[dst[i]], w[i]);
}

__global__ void k_dinv(float* __restrict__ deg, int n) {
    int i = blockIdx.x * blockDim.x + threadIdx.x;
    if (i < n) {
        float d = deg[i];
        deg[i] = (d > 0.0f) ? rsqrtf(d) : 0.0f;
    }
}

// ---------------------------------------------------------------- WMMA GEMM
// H[M,N] = A[M,128] * W[128,N].  N multiple of 64, M multiple of 16 handled by guard.
// Block: 256 threads = 8 waves; blockIdx.y selects a 64-col group; the 128x64 W
// slab (32KB) is staged once into LDS (async global->LDS when available) as
// K-pairs so each B fragment is one ds_load_b64.  Each wave then runs a 16-row
// tile with V_WMMA_F32_16X16X4_F32 (4 accumulators = 64 cols).
//
// A 16x4 layout:  lane l -> row m = l&15; lane-half h = l>>4; vgpr v holds K = kk + 2h + v
// B 4x16 layout:  lane l -> col n = l&15; vgpr v holds K = kk + 2h + v
// C/D 16x16:      vgpr i -> row = i + 8h, col = n
__global__ void __launch_bounds__(256)
k_gemm_wmma(const float* __restrict__ A, const float* __restrict__ W,
            float* __restrict__ H, int M, int N) {
    __shared__ float2 wlds[64 * 64];           // [kpair][col] => {W[2kp][c], W[2kp+1][c]}

    const int col0 = blockIdx.y * 64;
    const int t    = threadIdx.x;

    // ---- stage W slab: 4096 float2 pairs, 16 pairs per thread
#pragma unroll
    for (int i = 0; i < 16; ++i) {
        int p  = t + i * 256;                  // pair index 0..4095
        int kp = p >> 6;
        int c  = p & 63;
        const float* g0 = W + (size_t)(2 * kp) * N + col0 + c;
        const float* g1 = g0 + N;
        float* l = (float*)&wlds[p];
#if HAVE_ASYNC_LDS
        __builtin_amdgcn_global_load_async_to_lds_b32((as1_int*)g0, (as3_int*)(l + 0), 0, 0);
        __builtin_amdgcn_global_load_async_to_lds_b32((as1_int*)g1, (as3_int*)(l + 1), 0, 0);
#else
        l[0] = *g0;
        l[1] = *g1;
#endif
    }
#if HAVE_ASYNC_LDS
#if defined(__has_builtin) && __has_builtin(__builtin_amdgcn_s_wait_asynccnt)
    __builtin_amdgcn_s_wait_asynccnt(0);
#else
    asm volatile("s_wait_asynccnt 0" ::: "memory");
#endif
#endif
    __syncthreads();

    const int wave    = t >> 5;
    const int lane    = t & 31;
    const int rowTile = blockIdx.x * 8 + wave;
    if (rowTile * 16 >= M) return;             // wave-uniform exit after barrier

    const int row0 = rowTile * 16;
    const int m    = lane & 15;
    const int half = lane >> 4;

    v8f acc0 = {}, acc1 = {}, acc2 = {}, acc3 = {};

    const float*  __restrict__ arow  = A + (size_t)(row0 + m) * 128 + 2 * half;
    const float2* __restrict__ bbase = wlds + (size_t)half * 64 + m;

#pragma unroll 4
    for (int kk = 0; kk < 128; kk += 4) {
        v2f a;
        a.x = arow[kk + 0];
        a.y = arow[kk + 1];

        const float2* bp = bbase + (kk >> 1) * 64;   // kp = kk/2 + half
        float2 f0 = bp[0];
        float2 f1 = bp[16];
        float2 f2 = bp[32];
        float2 f3 = bp[48];
        v2f b0 = {f0.x, f0.y};
        v2f b1 = {f1.x, f1.y};
        v2f b2 = {f2.x, f2.y};
        v2f b3 = {f3.x, f3.y};

        acc0 = __builtin_amdgcn_wmma_f32_16x16x4_f32(false, a, false, b0, (short)0, acc0, false, false);
        acc1 = __builtin_amdgcn_wmma_f32_16x16x4_f32(false, a, false, b1, (short)0, acc1, false, false);
        acc2 = __builtin_amdgcn_wmma_f32_16x16x4_f32(false, a, false, b2, (short)0, acc2, false, false);
        acc3 = __builtin_amdgcn_wmma_f32_16x16x4_f32(false, a, false, b3, (short)0, acc3, false, false);
    }

#pragma unroll
    for (int i = 0; i < 8; ++i) {
        int r = row0 + i + 8 * half;
        float* __restrict__ hp = H + (size_t)r * N + col0 + m;
        hp[0]  = acc0[i];
        hp[16] = acc1[i];
        hp[32] = acc2[i];
        hp[48] = acc3[i];
    }
}

// ----------------------------------------------- self-loop + bias initializer
// out[i,c] = h[i,c] * dinv[i]^2 + bias[c]
__global__ void k_self_bias(const float* __restrict__ h, const float* __restrict__ dinv,
                            const float* __restrict__ bias, float* __restrict__ out,
                            int n, int c) {
    int cq  = c >> 2;
    int idx = blockIdx.x * blockDim.x + threadIdx.x;
    if (idx >= n * cq) return;
    int node = idx / cq;
    int ch   = (idx - node * cq) * 4;
    float dv = dinv[node];
    float s  = dv * dv;
    float4 hv = *(const float4*)(h + (size_t)node * c + ch);
    float4 o;
    o.x = hv.x * s + bias[ch + 0];
    o.y = hv.y * s + bias[ch + 1];
    o.z = hv.z * s + bias[ch + 2];
    o.w = hv.w * s + bias[ch + 3];
    *(float4*)(out + (size_t)node * c + ch) = o;
}

// ---------------------------------------------------------------- edge scatter
// out[dst] += h[src] * (dinv[src]*w*dinv[dst]); float4 per thread, wave ~ edge.
__global__ void k_scatter(const float* __restrict__ h, const float* __restrict__ dinv,
                          const int* __restrict__ src, const int* __restrict__ dst,
                          const float* __restrict__ w, float* __restrict__ out,
                          int e, int c) {
    int cq = c >> 2;
    long long idx = (long long)blockIdx.x * blockDim.x + threadIdx.x;
    if (idx >= (long long)e * cq) return;
    int ed = (int)(idx / cq);
    int ch = (int)(idx - (long long)ed * cq) * 4;
    int s = src[ed];
    int d = dst[ed];
    float norm = dinv[s] * w[ed] * dinv[d];
    float4 hv = *(const float4*)(h + (size_t)s * c + ch);
    float* op = out + (size_t)d * c + ch;
    atomicAdd(op + 0, hv.x * norm);
    atomicAdd(op + 1, hv.y * norm);
    atomicAdd(op + 2, hv.z * norm);
    atomicAdd(op + 3, hv.w * norm);
}

__global__ void k_relu(float* __restrict__ x, int n) {
    int i = blockIdx.x * blockDim.x + threadIdx.x;
    if (i < n) x[i] = fmaxf(x[i], 0.0f);
}

// ---------------------------------------------------------------- launcher
static inline int cdiv(long long a, int b) { return (int)((a + b - 1) / b); }

extern "C" void kernel_launch(void* const* d_in, const int* in_sizes, int n_in,
                              void* d_out, int out_size, void* d_ws, size_t ws_size,
                              hipStream_t stream) {
    const float* x   = (const float*)d_in[0];
    const int*   ei  = (const int*)  d_in[1];   // (2,E) flat: [src row | dst row]
    const float* ew  = (const float*)d_in[2];
    const float* W1  = (const float*)d_in[3];
    const float* b1  = (const float*)d_in[4];
    const float* W2  = (const float*)d_in[5];
    const float* b2  = (const float*)d_in[6];
    const float* W3  = (const float*)d_in[7];
    const float* b3  = (const float*)d_in[8];
    float* out = (float*)d_out;

    const int N = 100000;
    const int E = in_sizes[2];          // edge_weight element count
    const int CH = 128, CO = 64;
    const int* src = ei;
    const int* dst = ei + E;

    // workspace layout: dinv | tmp (N x 128) | act (N x 128)
    char* ws = (char*)d_ws;
    size_t off = 0;
    float* dinv = (float*)(ws + off); off += ((size_t)N * 4 + 255) & ~(size_t)255;
    float* tmp  = (float*)(ws + off); off += (size_t)N * CH * 4;
    float* act  = (float*)(ws + off);

    const int T = 256;

    // degree / normalization
    k_init_deg <<<cdiv(N, T), T, 0, stream>>>(dinv, N);
    k_accum_deg<<<cdiv(E, T), T, 0, stream>>>(dinv, dst, ew, E);
    k_dinv     <<<cdiv(N, T), T, 0, stream>>>(dinv, N);

    const int rowTiles  = N / 16;                 // 6250
    const dim3 gemmGrid128(cdiv(rowTiles, 8), 2); // 2 column groups for 128 cols
    const dim3 gemmGrid64 (cdiv(rowTiles, 8), 1); // 1 column group for 64 cols

    // ---- layer 1: x -> tmp -> act (relu)
    k_gemm_wmma<<<gemmGrid128, T, 0, stream>>>(x, W1, tmp, N, CH);
    k_self_bias<<<cdiv((long long)N * (CH/4), T), T, 0, stream>>>(tmp, dinv, b1, act, N, CH);
    k_scatter  <<<cdiv((long long)E * (CH/4), T), T, 0, stream>>>(tmp, dinv, src, dst, ew, act, E, CH);
    k_relu     <<<cdiv((long long)N * CH, T), T, 0, stream>>>(act, N * CH);

    // ---- layer 2: act -> tmp -> act (relu)
    k_gemm_wmma<<<gemmGrid128, T, 0, stream>>>(act, W2, tmp, N, CH);
    k_self_bias<<<cdiv((long long)N * (CH/4), T), T, 0, stream>>>(tmp, dinv, b2, act, N, CH);
    k_scatter  <<<cdiv((long long)E * (CH/4), T), T, 0, stream>>>(tmp, dinv, src, dst, ew, act, E, CH);
    k_relu     <<<cdiv((long long)N * CH, T), T, 0, stream>>>(act, N * CH);

    // ---- layer 3: act -> tmp -> out (no relu)
    k_gemm_wmma<<<gemmGrid64, T, 0, stream>>>(act, W3, tmp, N, CO);
    k_self_bias<<<cdiv((long long)N * (CO/4), T), T, 0, stream>>>(tmp, dinv, b3, out, N, CO);
    k_scatter  <<<cdiv((long long)E * (CO/4), T), T, 0, stream>>>(tmp, dinv, src, dst, ew, out, E, CO);
}